// TripletLoss_23880018166111
// MI455X (gfx1250) — compile-verified
//
#include <hip/hip_runtime.h>
#include <hip/hip_bf16.h>
#include <math.h>

// ---------------------------------------------------------------------------
// Triplet margin loss (B=8192, D=1024, 751 classes), MI455X / gfx1250.
//
// Phase 1: per-anchor first-positive / first-negative label scan (LDS-cached).
// Phase 2: per-wave, 16 anchors; squared-diff row sums reduced on the fp32
//          matrix pipe via V_WMMA_F32_16X16X4_F32 with an all-ones B matrix
//          (D[m][n] += sum_k A[m][k]); exact fp32, layout-robust.
// Phase 3: deterministic block-tree + final-tree reduction (no float atomics
//          -> bitwise-stable across graph replays).
// ---------------------------------------------------------------------------

typedef __attribute__((ext_vector_type(2))) float v2f;
typedef __attribute__((ext_vector_type(4))) float v4f;
typedef __attribute__((ext_vector_type(8))) float v8f;

#define TL_B      8192
#define TL_D      1024
#define TL_MARGIN 0.3f
#define TL_EPS    1e-6f
#define LOSS_BLOCKS 64   // 64 blocks * 8 waves * 16 anchors = 8192 anchors

// --------------------------- Phase 1: index search --------------------------
__global__ __launch_bounds__(256) void tl_idx_kernel(const int* __restrict__ labels,
                                                     int* __restrict__ posIdx,
                                                     int* __restrict__ negIdx,
                                                     int* __restrict__ validArr) {
    __shared__ int lab[TL_B];            // 32 KB of 320 KB WGP LDS
    const int tid = threadIdx.x;
    for (int j = tid; j < TL_B; j += 256) lab[j] = labels[j];
    __syncthreads();

    const int i  = blockIdx.x * 256 + tid;
    const int li = lab[i];
    int p = -1, n = -1;
    for (int j = 0; j < TL_B; ++j) {
        const int lj = lab[j];
        if (p < 0 && lj == li && j != i) p = j;
        if (n < 0 && lj != li)           n = j;
        if ((p >= 0) && (n >= 0)) break; // expected exit after ~750 iters (pos)
    }
    posIdx[i]   = (p < 0) ? 0 : p;       // argmax(all-False) == 0 in reference
    negIdx[i]   = (n < 0) ? 0 : n;
    validArr[i] = ((p >= 0) && (n >= 0)) ? 1 : 0;
}

// ----------------------- Phase 2: distances via WMMA ------------------------
__global__ __launch_bounds__(256) void tl_loss_kernel(const float* __restrict__ feat,
                                                      const int* __restrict__ posIdx,
                                                      const int* __restrict__ negIdx,
                                                      const int* __restrict__ validArr,
                                                      float* __restrict__ partialLoss,
                                                      float* __restrict__ partialCnt) {
    __shared__ float s_ap[8][16];
    __shared__ float s_an[8][16];
    __shared__ float bl[256];
    __shared__ float bc[256];

    const int tid  = threadIdx.x;
    const int w    = tid >> 5;           // wave in block (0..7)
    const int lane = tid & 31;
    const int half = lane >> 4;          // lane half selects K sub-columns
    const int m    = lane & 15;          // A-matrix row (ISA: M = lane % 16)
    const int g    = blockIdx.x * 8 + w; // 16-anchor group handled by this wave
    const int anchor = g * 16 + m;

    const int pi = posIdx[anchor];
    const int ni = negIdx[anchor];

    const float* A = feat + (size_t)anchor * TL_D;
    const float* P = feat + (size_t)pi     * TL_D;
    const float* N = feat + (size_t)ni     * TL_D;

    v8f cap = {0.f, 0.f, 0.f, 0.f, 0.f, 0.f, 0.f, 0.f};
    v8f can = {0.f, 0.f, 0.f, 0.f, 0.f, 0.f, 0.f, 0.f};
    const v2f ones = {1.f, 1.f};         // all-ones B: row sums, layout-proof

    const int cofs = half * 4;           // half 0 -> cols kb..kb+3, half 1 -> kb+4..kb+7
    for (int kb = 0; kb < TL_D; kb += 8) {
        const int c0 = kb + cofs;
        const v4f av = *(const v4f*)(A + c0);   // global_load_b128, 16B aligned
        const v4f pv = *(const v4f*)(P + c0);
        const v4f nv = *(const v4f*)(N + c0);

        const v4f dp  = av - pv + TL_EPS;       // torch: ||a - p + eps||_2
        const v4f dn  = av - nv + TL_EPS;
        const v4f dp2 = dp * dp;
        const v4f dn2 = dn * dn;

        const v2f aap0 = {dp2[0], dp2[1]};
        const v2f aap1 = {dp2[2], dp2[3]};
        const v2f aan0 = {dn2[0], dn2[1]};
        const v2f aan1 = {dn2[2], dn2[3]};

        // D = A x ones + C  ->  C[m][*] accumulates the squared-diff row sum.
        cap = __builtin_amdgcn_wmma_f32_16x16x4_f32(false, aap0, false, ones,
                                                    (short)0, cap, false, false);
        cap = __builtin_amdgcn_wmma_f32_16x16x4_f32(false, aap1, false, ones,
                                                    (short)0, cap, false, false);
        can = __builtin_amdgcn_wmma_f32_16x16x4_f32(false, aan0, false, ones,
                                                    (short)0, can, false, false);
        can = __builtin_amdgcn_wmma_f32_16x16x4_f32(false, aan1, false, ones,
                                                    (short)0, can, false, false);
    }

    // C layout (ISA 7.12.2): lane 0 holds rows 0..7, col 0 in c[0..7];
    // lane 16 holds rows 8..15, col 0.
    if (lane == 0) {
#pragma unroll
        for (int r = 0; r < 8; ++r) { s_ap[w][r] = cap[r]; s_an[w][r] = can[r]; }
    }
    if (lane == 16) {
#pragma unroll
        for (int r = 0; r < 8; ++r) { s_ap[w][8 + r] = cap[r]; s_an[w][8 + r] = can[r]; }
    }
    __syncthreads();

    float loss = 0.f, cnt = 0.f;
    if (lane < 16) {
        const int   v   = validArr[anchor];
        const float dap = sqrtf(s_ap[w][lane]);
        const float dan = sqrtf(s_an[w][lane]);
        const float l   = dap - dan + TL_MARGIN;
        if (v) { cnt = 1.f; loss = (l > 0.f) ? l : 0.f; }
    }
    bl[tid] = loss;
    bc[tid] = cnt;
    __syncthreads();
    for (int s = 128; s > 0; s >>= 1) {
        if (tid < s) { bl[tid] += bl[tid + s]; bc[tid] += bc[tid + s]; }
        __syncthreads();
    }
    if (tid == 0) { partialLoss[blockIdx.x] = bl[0]; partialCnt[blockIdx.x] = bc[0]; }
}

// --------------------------- Phase 3: finalize ------------------------------
__global__ __launch_bounds__(64) void tl_finalize(const float* __restrict__ pl,
                                                  const float* __restrict__ pc,
                                                  float* __restrict__ out) {
    __shared__ float sl[64];
    __shared__ float sc[64];
    const int t = threadIdx.x;
    sl[t] = pl[t];
    sc[t] = pc[t];
    __syncthreads();
    for (int s = 32; s > 0; s >>= 1) {
        if (t < s) { sl[t] += sl[t + s]; sc[t] += sc[t + s]; }
        __syncthreads();
    }
    if (t == 0) out[0] = (sc[0] > 0.f) ? (sl[0] / sc[0]) : 0.f;
}

// ----------------------------------------------------------------------------
extern "C" void kernel_launch(void* const* d_in, const int* in_sizes, int n_in,
                              void* d_out, int out_size, void* d_ws, size_t ws_size,
                              hipStream_t stream) {
    const float* feat   = (const float*)d_in[0];
    const int*   labels = (const int*)d_in[1];
    float*       out    = (float*)d_out;

    char* ws = (char*)d_ws;                       // ~96.5 KB used, fully
    int*   posIdx = (int*)(ws);                   // rewritten every call
    int*   negIdx = (int*)(ws + (size_t)TL_B * 4);
    int*   valid  = (int*)(ws + (size_t)TL_B * 8);
    float* pl     = (float*)(ws + (size_t)TL_B * 12);
    float* pc     = (float*)(ws + (size_t)TL_B * 12 + LOSS_BLOCKS * 4);

    tl_idx_kernel<<<TL_B / 256, 256, 0, stream>>>(labels, posIdx, negIdx, valid);
    tl_loss_kernel<<<LOSS_BLOCKS, 256, 0, stream>>>(feat, posIdx, negIdx, valid, pl, pc);
    tl_finalize<<<1, 64, 0, stream>>>(pl, pc, out);
}